// PredictYolov1_1_84293028151779
// MI455X (gfx1250) — compile-verified
//
#include <hip/hip_runtime.h>
#include <hip/hip_bf16.h>
#include <stdint.h>

// ---------------- problem constants ----------------
#define NB        4
#define NCLS      80
#define NCH       85          // 1 + 80 + 4
#define HW        676         // 26*26
#define GRID      26
#define N_ALL     2704        // NB * HW
#define N_PAD     4096        // sort padding (power of two)
#define N_WORDS   85          // ceil(N_ALL / 32)
#define N_ITILES  169         // N_ALL / 16
#define THR_CONF  0.02f
#define THR_NMS   0.5f
#define IMG_SZ    416.0f

typedef float v2f __attribute__((ext_vector_type(2)));
typedef float v8f __attribute__((ext_vector_type(8)));

// =====================================================================
// Kernel 1: decode. One thread per box.
// out layout: [ids N][boxes 4N][labels N][scores N][keep N]
// ws: keys (masked scores) [N], boxOff [4N]
// =====================================================================
__global__ __launch_bounds__(256)
void decode_kernel(const float* __restrict__ in, float* __restrict__ out,
                   float* __restrict__ keys, float* __restrict__ boxOff) {
    int n = blockIdx.x * blockDim.x + threadIdx.x;
    if (n >= N_ALL) return;
    int b = n / HW;
    int s = n - b * HW;
    const float* p = in + (size_t)(b * NCH) * HW + s;

    // objectness
    float pconf = 1.f / (1.f + expf(-p[0]));

    // softmax over 80 classes: need max, argmax, sum(exp(L - max))
    float maxL = -3.4e38f; int arg = 0;
    for (int c = 0; c < NCLS; ++c) {
        float L = p[(size_t)(1 + c) * HW];
        if (L > maxL) { maxL = L; arg = c; }
    }
    float sumE = 0.f;
    for (int c = 0; c < NCLS; ++c)
        sumE += expf(p[(size_t)(1 + c) * HW] - maxL);
    float clsConf = 1.f / sumE;           // exp(max)/sum(exp)
    float score = clsConf * pconf;

    // box decode
    float tx = p[(size_t)(1 + NCLS + 0) * HW];
    float ty = p[(size_t)(1 + NCLS + 1) * HW];
    float tw = p[(size_t)(1 + NCLS + 2) * HW];
    float th = p[(size_t)(1 + NCLS + 3) * HW];
    int gx = s % GRID, gy = s / GRID;
    float x = (1.f / (1.f + expf(-tx)) + (float)gx) / (float)GRID;
    float y = (1.f / (1.f + expf(-ty)) + (float)gy) / (float)GRID;
    float w = expf(tw) / IMG_SZ;
    float h = expf(th) / IMG_SZ;
    float bl = fminf(fmaxf(x - 0.5f * w, 0.f), 1.f);
    float bt = fminf(fmaxf(y - 0.5f * h, 0.f), 1.f);
    float br = fminf(fmaxf(x + 0.5f * w, 0.f), 1.f);
    float bb = fminf(fmaxf(y + 0.5f * h, 0.f), 1.f);

    int label = arg + 1;

    out[n] = (float)b;                       // ids_batch
    out[N_ALL + 4 * n + 0] = bl;             // boxes
    out[N_ALL + 4 * n + 1] = bt;
    out[N_ALL + 4 * n + 2] = br;
    out[N_ALL + 4 * n + 3] = bb;
    out[5 * N_ALL + n] = (float)label;       // labels
    out[6 * N_ALL + n] = score;              // scores

    float sm = (score > THR_CONF) ? score : 0.f;
    keys[n] = sm;
    float off = (float)(b * (NCLS + 2) + label) * 2.0f;
    boxOff[4 * n + 0] = bl + off;
    boxOff[4 * n + 1] = bt + off;
    boxOff[4 * n + 2] = br + off;
    boxOff[4 * n + 3] = bb + off;
}

// =====================================================================
// Kernel 2: bitonic sort of 4096 (negated score, index) pairs in LDS.
// Ascending on (-score, idx) == descending score with stable tiebreak.
// Emits sorted scores, sorted original indices, gathered offset boxes.
// =====================================================================
__global__ __launch_bounds__(1024)
void sort_kernel(const float* __restrict__ keys, const float* __restrict__ boxOff,
                 float* __restrict__ sScore, int* __restrict__ sIdx,
                 float* __restrict__ sBox) {
    __shared__ float k[N_PAD];
    __shared__ int   id[N_PAD];
    int tid = threadIdx.x;
    for (int t = tid; t < N_PAD; t += 1024) {
        k[t]  = (t < N_ALL) ? -keys[t] : 3.0e38f;   // pads sort to the end
        id[t] = t;
    }
    __syncthreads();
    for (int kk = 2; kk <= N_PAD; kk <<= 1) {
        for (int j = kk >> 1; j > 0; j >>= 1) {
            for (int t = tid; t < N_PAD; t += 1024) {
                int l = t ^ j;
                if (l > t) {
                    bool up = ((t & kk) == 0);
                    float ka = k[t], kb = k[l];
                    int   ia = id[t], ib = id[l];
                    // element l strictly before element t in ascending order?
                    bool lessBA = (kb < ka) || (kb == ka && ib < ia);
                    if (lessBA == up) {
                        k[t] = kb; k[l] = ka;
                        id[t] = ib; id[l] = ia;
                    }
                }
            }
            __syncthreads();
        }
    }
    for (int t = tid; t < N_ALL; t += 1024) {
        int o = id[t];
        sScore[t] = -k[t];
        sIdx[t] = o;
        sBox[4 * t + 0] = boxOff[4 * o + 0];
        sBox[4 * t + 1] = boxOff[4 * o + 1];
        sBox[4 * t + 2] = boxOff[4 * o + 2];
        sBox[4 * t + 3] = boxOff[4 * o + 3];
    }
}

// =====================================================================
// Kernel 3: suppression bitmask. One wave32 per 16-row x 32-col tile.
// areaSum[i][j] = area_i + area_j computed with V_WMMA_F32_16X16X4_F32:
//   A(16x4) rows = [area_i, 1, 0, 0], B(4x16) = [1; area_j; 0; 0]
// Bits packed via wave32 ballot into supp[row][word].
// =====================================================================
__global__ __launch_bounds__(32)
void supp_kernel(const float* __restrict__ sbox, uint32_t* __restrict__ supp) {
    int blk = blockIdx.x;
    int bi = blk % N_ITILES;       // row tile
    int wj = blk / N_ITILES;       // column word index
    int lane = threadIdx.x;
    int i_base = bi * 16;
    int j_base = wj * 32;

    __shared__ float rbox[16][4];
    if (lane < 16) {
        int r = i_base + lane;     // always < N_ALL
        rbox[lane][0] = sbox[4 * r + 0];
        rbox[lane][1] = sbox[4 * r + 1];
        rbox[lane][2] = sbox[4 * r + 2];
        rbox[lane][3] = sbox[4 * r + 3];
    }
    __syncthreads();

    // column boxes: this lane's column within each 16-wide half-tile
    int jc0 = j_base + (lane & 15);
    int jc1 = jc0 + 16;
    bool v0 = jc0 < N_ALL, v1 = jc1 < N_ALL;
    float c0l = 0, c0t = 0, c0r = 0, c0b = 0;
    float c1l = 0, c1t = 0, c1r = 0, c1b = 0;
    if (v0) { c0l = sbox[4*jc0+0]; c0t = sbox[4*jc0+1]; c0r = sbox[4*jc0+2]; c0b = sbox[4*jc0+3]; }
    if (v1) { c1l = sbox[4*jc1+0]; c1t = sbox[4*jc1+1]; c1r = sbox[4*jc1+2]; c1b = sbox[4*jc1+3]; }
    float areaJ0 = (c0r - c0l) * (c0b - c0t);
    float areaJ1 = (c1r - c1l) * (c1b - c1t);

    // A operand (16x4 f32, 2 VGPRs): v0 = K0(l0-15)/K2(l16-31), v1 = K1/K3
    float areaI = 0.f;
    if (lane < 16)
        areaI = (rbox[lane][2] - rbox[lane][0]) * (rbox[lane][3] - rbox[lane][1]);
    v2f A;  A.x  = (lane < 16) ? areaI : 0.f;   A.y  = (lane < 16) ? 1.f : 0.f;
    v2f B0; B0.x = (lane < 16) ? 1.f   : 0.f;   B0.y = (lane < 16) ? areaJ0 : 0.f;
    v2f B1; B1.x = (lane < 16) ? 1.f   : 0.f;   B1.y = (lane < 16) ? areaJ1 : 0.f;

    v8f zero = {};
    // D[m][n] = area_i[m] + area_j[n]
    v8f S0 = __builtin_amdgcn_wmma_f32_16x16x4_f32(false, A, false, B0,
                                                   (short)0, zero, false, false);
    v8f S1 = __builtin_amdgcn_wmma_f32_16x16x4_f32(false, A, false, B1,
                                                   (short)0, zero, false, false);

    // D layout: VGPR r -> lanes 0-15: (M=r, N=lane); lanes 16-31: (M=r+8, N=lane-16)
    for (int r = 0; r < 8; ++r) {
        int rowLocal = r + ((lane >= 16) ? 8 : 0);
        int row = i_base + rowLocal;
        float bl = rbox[rowLocal][0], bt = rbox[rowLocal][1];
        float br = rbox[rowLocal][2], bb = rbox[rowLocal][3];

        // half 0
        float iw0 = fmaxf(fminf(br, c0r) - fmaxf(bl, c0l), 0.f);
        float ih0 = fmaxf(fminf(bb, c0b) - fmaxf(bt, c0t), 0.f);
        float inter0 = iw0 * ih0;
        float den0 = S0[r] - inter0 + 1e-9f;
        bool p0 = v0 && (jc0 > row) && (inter0 > THR_NMS * den0);

        // half 1
        float iw1 = fmaxf(fminf(br, c1r) - fmaxf(bl, c1l), 0.f);
        float ih1 = fmaxf(fminf(bb, c1b) - fmaxf(bt, c1t), 0.f);
        float inter1 = iw1 * ih1;
        float den1 = S1[r] - inter1 + 1e-9f;
        bool p1 = v1 && (jc1 > row) && (inter1 > THR_NMS * den1);

        uint32_t bal0 = __builtin_amdgcn_ballot_w32(p0);
        uint32_t bal1 = __builtin_amdgcn_ballot_w32(p1);
        if (lane == 0) {
            supp[(size_t)(i_base + r)     * N_WORDS + wj] = (bal0 & 0xFFFFu) | (bal1 << 16);
            supp[(size_t)(i_base + r + 8) * N_WORDS + wj] = (bal0 >> 16)     | (bal1 & 0xFFFF0000u);
        }
    }
}

// =====================================================================
// Kernel 4: serial greedy scan, single wave32.
// removed[] bitmask lives in LDS; scores/indices preloaded into LDS.
// keep[sortedIdx[i]] = flag, flag = keep0[i] && !removed[i]
// =====================================================================
__global__ __launch_bounds__(32)
void nms_scan_kernel(const float* __restrict__ sScore, const int* __restrict__ sIdx,
                     const uint32_t* __restrict__ supp, float* __restrict__ keepOut) {
    __shared__ uint32_t removed[N_WORDS];
    __shared__ float sc[N_ALL];
    __shared__ int   idx[N_ALL];
    int lane = threadIdx.x;
    for (int t = lane; t < N_WORDS; t += 32) removed[t] = 0u;
    for (int t = lane; t < N_ALL; t += 32) { sc[t] = sScore[t]; idx[t] = sIdx[t]; }
    __syncthreads();

    for (int i = 0; i < N_ALL; ++i) {
        uint32_t rw = removed[i >> 5];
        bool flag = (sc[i] > THR_CONF) && (((rw >> (i & 31)) & 1u) == 0u);
        if (flag) {
            for (int w = lane; w < N_WORDS; w += 32)
                removed[w] |= supp[(size_t)i * N_WORDS + w];
        }
        if (lane == 0) keepOut[idx[i]] = flag ? 1.f : 0.f;
        __syncthreads();
    }
}

// =====================================================================
extern "C" void kernel_launch(void* const* d_in, const int* in_sizes, int n_in,
                              void* d_out, int out_size, void* d_ws, size_t ws_size,
                              hipStream_t stream) {
    (void)in_sizes; (void)n_in; (void)out_size; (void)ws_size;
    const float* in = (const float*)d_in[0];
    float* out = (float*)d_out;

    // workspace layout (all 4-byte types, base is aligned)
    float*    keys   = (float*)d_ws;               // N_ALL
    float*    boxOff = keys + N_ALL;               // 4*N_ALL
    float*    sScore = boxOff + 4 * N_ALL;         // N_ALL
    int*      sIdx   = (int*)(sScore + N_ALL);     // N_ALL
    float*    sBox   = (float*)(sIdx + N_ALL);     // 4*N_ALL
    uint32_t* supp   = (uint32_t*)(sBox + 4 * N_ALL); // N_ALL * N_WORDS

    decode_kernel<<<(N_ALL + 255) / 256, 256, 0, stream>>>(in, out, keys, boxOff);
    sort_kernel<<<1, 1024, 0, stream>>>(keys, boxOff, sScore, sIdx, sBox);
    supp_kernel<<<N_ITILES * N_WORDS, 32, 0, stream>>>(sBox, supp);
    nms_scan_kernel<<<1, 32, 0, stream>>>(sScore, sIdx, supp, out + 7 * N_ALL);
}